// ZIGNNExpert_20538533609923
// MI455X (gfx1250) — compile-verified
//
#include <hip/hip_runtime.h>
#include <hip/hip_bf16.h>
#include <math.h>

#define HID   128
#define HEADS 4
#define NEG_SLOPE 0.2f

typedef __attribute__((ext_vector_type(2))) float v2f;
typedef __attribute__((ext_vector_type(8))) float v8f;

__device__ __forceinline__ float gelu_tanh(float x) {
  const float k0 = 0.7978845608028654f; // sqrt(2/pi)
  float x3 = x * x * x;
  return 0.5f * x * (1.0f + tanhf(k0 * (x + 0.044715f * x3)));
}

__device__ __forceinline__ float wave_reduce_add(float v) {
  v += __shfl_xor(v, 1, 32);
  v += __shfl_xor(v, 2, 32);
  v += __shfl_xor(v, 4, 32);
  v += __shfl_xor(v, 8, 32);
  v += __shfl_xor(v, 16, 32);
  return v;
}

// monotonic float atomic-max: signed-int max for >=0, unsigned min for <0
__device__ __forceinline__ void atomic_max_f32(float* addr, float v) {
  if (v >= 0.0f) atomicMax((int*)addr, __float_as_int(v));
  else           atomicMin((unsigned int*)addr, __float_as_uint(v));
}

__global__ void fill_kernel(float* __restrict__ p, float v, int n) {
  int i = blockIdx.x * blockDim.x + threadIdx.x;
  if (i < n) p[i] = v;
}

// ---------------------------------------------------------------------------
// Y[M x Ncols] = X[M x K] @ W[K x Ncols] + bias, optional GELU.
// One wave32 per 16x16 tile; K-loop of V_WMMA_F32_16X16X4_F32 (fp32 matrix op).
// A frag (16x4 f32): lanes 0-15 hold M=lane, VGPR0/1 = K=0/1; lanes 16-31 K=2/3.
// B frag (4x16 f32): lane = N, VGPR0/1 = K rows (0,1 | 2,3 by lane half).
// C/D (16x16 f32): 8 VGPRs, lanes 0-15 M=r, lanes 16-31 M=r+8.
// ---------------------------------------------------------------------------
__global__ void wmma_gemm_f32(const float* __restrict__ X, const float* __restrict__ W,
                              const float* __restrict__ bias, float* __restrict__ Y,
                              int Mrows, int K, int Ncols, int act) {
  const int tilesN = Ncols >> 4;
  const int tiles  = (Mrows >> 4) * tilesN;
  const int tile   = blockIdx.x * blockDim.y + threadIdx.y; // wave-uniform
  if (tile >= tiles) return;                                // whole-wave exit (EXEC stays all-1 for WMMA)
  const int tM = tile / tilesN, tN = tile % tilesN;
  const int lane = threadIdx.x;
  const int lm = lane & 15;
  const int lk = (lane >> 4) << 1; // 0 or 2

  const float* __restrict__ xrow = X + (size_t)(tM * 16 + lm) * K;
  const float* __restrict__ wcol = W + (size_t)(tN * 16 + lm);
  v8f c = {};
  for (int k = 0; k < K; k += 4) {
    v2f a, b;
    a.x = xrow[k + lk];
    a.y = xrow[k + lk + 1];
    const float* wr = wcol + (size_t)(k + lk) * Ncols;
    b.x = wr[0];
    b.y = wr[Ncols];
    c = __builtin_amdgcn_wmma_f32_16x16x4_f32(false, a, false, b, (short)0, c, false, false);
  }
  const int col = tN * 16 + lm;
  const float bv = bias ? bias[col] : 0.0f;
  const int rbase = tM * 16 + ((lane >> 4) << 3);
  #pragma unroll
  for (int r = 0; r < 8; ++r) {
    float v = c[r] + bv;
    if (act == 1) v = gelu_tanh(v);
    Y[(size_t)(rbase + r) * Ncols + col] = v;
  }
}

// one wave per row of 128; float4 per lane, shfl reductions
__global__ void layernorm_kernel(const float* __restrict__ X, const float* __restrict__ g,
                                 const float* __restrict__ b, float* __restrict__ Y, int rows) {
  int row = blockIdx.x * blockDim.y + threadIdx.y;
  if (row >= rows) return;
  int lane = threadIdx.x;
  float4 v = ((const float4*)(X + (size_t)row * HID))[lane];
  float s = v.x + v.y + v.z + v.w;
  float q = v.x * v.x + v.y * v.y + v.z * v.z + v.w * v.w;
  s = wave_reduce_add(s);
  q = wave_reduce_add(q);
  float mu  = s * (1.0f / HID);
  float var = q * (1.0f / HID) - mu * mu;
  float rs  = rsqrtf(var + 1e-5f);
  int c = lane * 4;
  float* y = Y + (size_t)row * HID;
  y[c + 0] = (v.x - mu) * rs * g[c + 0] + b[c + 0];
  y[c + 1] = (v.y - mu) * rs * g[c + 1] + b[c + 1];
  y[c + 2] = (v.z - mu) * rs * g[c + 2] + b[c + 2];
  y[c + 3] = (v.w - mu) * rs * g[c + 3] + b[c + 3];
}

// Pass A: one wave per edge. Gather xl[src], xr[dst] (b128/lane), leaky-relu,
// dot with att per head (8 lanes/head), write alpha[E,4] + atomic segment max.
__global__ void edge_alpha_kernel(const float* __restrict__ xl, const float* __restrict__ xr,
                                  const int* __restrict__ src, const int* __restrict__ dst,
                                  const float* __restrict__ att, float* __restrict__ alpha,
                                  float* __restrict__ amax, int E) {
  int e = blockIdx.x * blockDim.y + threadIdx.y;
  if (e >= E) return;
  int lane = threadIdx.x;
  int s = src[e], d = dst[e];
  float4 l4 = ((const float4*)(xl + (size_t)s * HID))[lane];
  float4 r4 = ((const float4*)(xr + (size_t)d * HID))[lane];
  float4 a4 = ((const float4*)att)[lane]; // att[h,c] flat: lane covers head lane>>3
  float ex0 = l4.x + r4.x, ex1 = l4.y + r4.y, ex2 = l4.z + r4.z, ex3 = l4.w + r4.w;
  ex0 = ex0 > 0.0f ? ex0 : ex0 * NEG_SLOPE;
  ex1 = ex1 > 0.0f ? ex1 : ex1 * NEG_SLOPE;
  ex2 = ex2 > 0.0f ? ex2 : ex2 * NEG_SLOPE;
  ex3 = ex3 > 0.0f ? ex3 : ex3 * NEG_SLOPE;
  float p = ex0 * a4.x + ex1 * a4.y + ex2 * a4.z + ex3 * a4.w;
  p += __shfl_xor(p, 1, 32);
  p += __shfl_xor(p, 2, 32);
  p += __shfl_xor(p, 4, 32);
  if ((lane & 7) == 0) {
    int hh = lane >> 3;
    alpha[(size_t)e * HEADS + hh] = p;
    atomic_max_f32(&amax[(size_t)d * HEADS + hh], p);
  }
}

// Pass B: ex = exp(alpha - amax[dst]) in place; denom[dst] += ex (f32 atomic)
__global__ void edge_softmax_kernel(const int* __restrict__ dst, float* __restrict__ alpha,
                                    const float* __restrict__ amax, float* __restrict__ denom,
                                    int EH) {
  int t = blockIdx.x * blockDim.x + threadIdx.x;
  if (t >= EH) return;
  int e = t >> 2, hh = t & 3;
  int d = dst[e];
  float ex = expf(alpha[t] - amax[(size_t)d * HEADS + hh]);
  alpha[t] = ex;
  unsafeAtomicAdd(&denom[(size_t)d * HEADS + hh], ex);
}

// Pass C: one wave per edge; w = ex/denom[dst]; out[dst] += xl[src]*w (4 f32 atomics/lane)
__global__ void edge_scatter_kernel(const float* __restrict__ xl, const int* __restrict__ src,
                                    const int* __restrict__ dst, const float* __restrict__ alpha,
                                    const float* __restrict__ denom, float* __restrict__ out,
                                    int E) {
  int e = blockIdx.x * blockDim.y + threadIdx.y;
  if (e >= E) return;
  int lane = threadIdx.x;
  int s = src[e], d = dst[e];
  int hh = lane >> 3;
  float w = alpha[(size_t)e * HEADS + hh] / (denom[(size_t)d * HEADS + hh] + 1e-16f);
  float4 v = ((const float4*)(xl + (size_t)s * HID))[lane];
  float* o = out + (size_t)d * HID + lane * 4;
  unsafeAtomicAdd(o + 0, v.x * w);
  unsafeAtomicAdd(o + 1, v.y * w);
  unsafeAtomicAdd(o + 2, v.z * w);
  unsafeAtomicAdd(o + 3, v.w * w);
}

// h += gelu(conv_out + conv_b)
__global__ void residual_gelu_kernel(float* __restrict__ h, const float* __restrict__ cv,
                                     const float* __restrict__ conv_b, int total) {
  int i = blockIdx.x * blockDim.x + threadIdx.x;
  if (i >= total) return;
  float zv = cv[i] + conv_b[i & (HID - 1)];
  h[i] += gelu_tanh(zv);
}

// logits = t1[N,64] @ pW2[64,1] + pb2 : wave per row, shfl-reduced dot
__global__ void prob_head_kernel(const float* __restrict__ t1, const float* __restrict__ w2,
                                 const float* __restrict__ b2, float* __restrict__ out, int rows) {
  int row = blockIdx.x * blockDim.y + threadIdx.y;
  if (row >= rows) return;
  int lane = threadIdx.x;
  const float* tr = t1 + (size_t)row * 64;
  float p = tr[lane] * w2[lane] + tr[lane + 32] * w2[lane + 32];
  p = wave_reduce_add(p);
  if (lane == 0) out[row] = p + b2[0];
}

extern "C" void kernel_launch(void* const* d_in, const int* in_sizes, int n_in,
                              void* d_out, int out_size, void* d_ws, size_t ws_size,
                              hipStream_t stream) {
  const float* x      = (const float*)d_in[0];
  const int*   ei     = (const int*)d_in[1];   // [2,E] (int32 under default JAX x64-off)
  const float* Win    = (const float*)d_in[2];
  const float* b_in   = (const float*)d_in[3];
  const float* ln_g   = (const float*)d_in[4];
  const float* ln_b   = (const float*)d_in[5];
  const float* Wl     = (const float*)d_in[6];
  const float* bl     = (const float*)d_in[7];
  const float* Wr     = (const float*)d_in[8];
  const float* br     = (const float*)d_in[9];
  const float* att    = (const float*)d_in[10];
  const float* conv_b = (const float*)d_in[11];
  const float* fn_g   = (const float*)d_in[12];
  const float* fn_b   = (const float*)d_in[13];
  const float* pW1    = (const float*)d_in[14];
  const float* pb1    = (const float*)d_in[15];
  const float* pW2    = (const float*)d_in[16];
  const float* pb2    = (const float*)d_in[17];
  const float* vW1    = (const float*)d_in[18];
  const float* vb1    = (const float*)d_in[19];
  const float* vW2    = (const float*)d_in[20];
  const float* vb2    = (const float*)d_in[21];

  const int N = in_sizes[0] / HID;
  const int E = in_sizes[1] / 2;
  const int* src = ei;
  const int* dst = ei + E;

  // workspace layout (fp32)
  float* ws    = (float*)d_ws;
  size_t NH    = (size_t)N * HID;
  float* h     = ws;                       // [N,128]
  float* z     = h     + NH;               // [N,128]
  float* xl    = z     + NH;               // [N,128]
  float* xr    = xl    + NH;               // [N,128]
  float* cv    = xr    + NH;               // [N,128] conv out
  float* alpha = cv    + NH;               // [E,4]  alpha -> ex (in place)
  float* amax  = alpha + (size_t)E * HEADS;// [N,4]
  float* denom = amax  + (size_t)N * HEADS;// [N,4]
  float* t1    = denom + (size_t)N * HEADS;// [N,64]

  float* volume = (float*)d_out;           // [N,128] first
  float* logits = volume + NH;             // [N,1]   second

  dim3 wblk(32, 8); // 8 wave32 per block

  auto gemm = [&](const float* X, const float* W, const float* bias, float* Y,
                  int M, int K, int Ncols, int act) {
    int tiles = (M / 16) * (Ncols / 16);
    wmma_gemm_f32<<<(tiles + 7) / 8, wblk, 0, stream>>>(X, W, bias, Y, M, K, Ncols, act);
  };

  // input projection: h = x @ Win + b_in
  gemm(x, Win, b_in, h, N, HID, HID, 0);

  for (int i = 0; i < 3; ++i) {
    layernorm_kernel<<<(N + 7) / 8, wblk, 0, stream>>>(h, ln_g + i * HID, ln_b + i * HID, z, N);
    gemm(z, Wl + (size_t)i * HID * HID, bl + i * HID, xl, N, HID, HID, 0);
    gemm(z, Wr + (size_t)i * HID * HID, br + i * HID, xr, N, HID, HID, 0);

    fill_kernel<<<(N * HEADS + 255) / 256, 256, 0, stream>>>(amax, -INFINITY, N * HEADS);
    fill_kernel<<<(N * HEADS + 255) / 256, 256, 0, stream>>>(denom, 0.0f, N * HEADS);
    fill_kernel<<<((int)NH + 255) / 256, 256, 0, stream>>>(cv, 0.0f, (int)NH);

    edge_alpha_kernel  <<<(E + 7) / 8, wblk, 0, stream>>>(xl, xr, src, dst, att + i * HID,
                                                          alpha, amax, E);
    edge_softmax_kernel<<<(E * HEADS + 255) / 256, 256, 0, stream>>>(dst, alpha, amax, denom,
                                                                     E * HEADS);
    edge_scatter_kernel<<<(E + 7) / 8, wblk, 0, stream>>>(xl, src, dst, alpha, denom, cv, E);

    residual_gelu_kernel<<<((int)NH + 255) / 256, 256, 0, stream>>>(h, cv, conv_b + i * HID,
                                                                    (int)NH);
  }

  // final norm + heads
  layernorm_kernel<<<(N + 7) / 8, wblk, 0, stream>>>(h, fn_g, fn_b, z, N);

  gemm(z, pW1, pb1, t1, N, HID, HID / 2, 1);            // gelu(z @ pW1 + pb1) -> [N,64]
  prob_head_kernel<<<(N + 7) / 8, wblk, 0, stream>>>(t1, pW2, pb2, logits, N);

  gemm(z, vW1, vb1, t1, N, HID, HID / 2, 1);            // gelu(z @ vW1 + vb1) -> [N,64]
  gemm(t1, vW2, vb2, volume, N, HID / 2, HID, 0);       // volume_emb [N,128]
}